// GATLayer_35854386987429
// MI455X (gfx1250) — compile-verified
//
#include <hip/hip_runtime.h>

typedef __attribute__((ext_vector_type(2))) float v2f;
typedef __attribute__((ext_vector_type(8))) float v8f;

#define IN_DIM 256
#define OUT_DIM 64
#define NEG_SLOPE 0.2f

// ---------------------------------------------------------------------------
// Kernel 1: h = x @ W  via V_WMMA_F32_16X16X4_F32.
// One wave computes a 16-row block of h across all 64 output columns
// (4 N-tiles), so x is streamed exactly once. K-loop: 256/4 = 64 WMMAs/tile.
// A layout (16x4 f32): lanes 0-15 -> K=0,1 in v0,v1 ; lanes 16-31 -> K=2,3.
// B layout (4x16 f32): lanes 0-15 -> rows K=0,1 ; lanes 16-31 -> rows K=2,3.
// C/D layout: VGPR v -> M = v (lanes 0-15) / v+8 (lanes 16-31), N = lane%16.
// ---------------------------------------------------------------------------
__global__ void gat_gemm_wmma(const float* __restrict__ x,
                              const float* __restrict__ W,
                              float* __restrict__ h, int nrows) {
  int wave = (int)((blockIdx.x * (size_t)blockDim.x + threadIdx.x) >> 5);
  int lane = threadIdx.x & 31;
  int row0 = wave * 16;
  if (row0 >= nrows) return;
  int half = lane >> 4;   // 0: K pair {0,1}, 1: K pair {2,3}
  int l16  = lane & 15;

  const float* arow = x + (size_t)(row0 + l16) * IN_DIM;
  v8f acc0 = {}, acc1 = {}, acc2 = {}, acc3 = {};

  for (int k = 0; k < IN_DIM; k += 4) {
    int ka = k + half * 2;
    v2f a;
    a.x = arow[ka];
    a.y = arow[ka + 1];
    const float* wr = W + (size_t)ka * OUT_DIM;   // rows ka, ka+1 of W
    v2f b0, b1, b2, b3;
    b0.x = wr[l16];          b0.y = wr[OUT_DIM + l16];
    b1.x = wr[16 + l16];     b1.y = wr[OUT_DIM + 16 + l16];
    b2.x = wr[32 + l16];     b2.y = wr[OUT_DIM + 32 + l16];
    b3.x = wr[48 + l16];     b3.y = wr[OUT_DIM + 48 + l16];
    acc0 = __builtin_amdgcn_wmma_f32_16x16x4_f32(false, a, false, b0, (short)0, acc0, false, false);
    acc1 = __builtin_amdgcn_wmma_f32_16x16x4_f32(false, a, false, b1, (short)0, acc1, false, false);
    acc2 = __builtin_amdgcn_wmma_f32_16x16x4_f32(false, a, false, b2, (short)0, acc2, false, false);
    acc3 = __builtin_amdgcn_wmma_f32_16x16x4_f32(false, a, false, b3, (short)0, acc3, false, false);
  }

#pragma unroll
  for (int v = 0; v < 8; ++v) {
    int row = row0 + v + 8 * half;
    float* orow = h + (size_t)row * OUT_DIM + l16;
    orow[0]  = acc0[v];
    orow[16] = acc1[v];
    orow[32] = acc2[v];
    orow[48] = acc3[v];
  }
}

// ---------------------------------------------------------------------------
// Kernel 2: init out=0, m=-inf, denom=0   (harness poisons d_out/d_ws)
// ---------------------------------------------------------------------------
__global__ void gat_init(float* __restrict__ out, float* __restrict__ m,
                         float* __restrict__ denom, int n) {
  int t = blockIdx.x * blockDim.x + threadIdx.x;
  if (t >= n * OUT_DIM) return;
  out[t] = 0.0f;
  if ((t & (OUT_DIM - 1)) == 0) {
    int node = t >> 6;
    m[node] = __int_as_float(0xff800000);   // -inf
    denom[node] = 0.0f;
  }
}

// ---------------------------------------------------------------------------
// Kernel 3: per-node attention scores  s_src = h@a[:64], s_dst = h@a[64:]
// h is L2-resident (25.6 MB << 192 MB L2) so scalar reloads are cheap.
// ---------------------------------------------------------------------------
__global__ void gat_scores(const float* __restrict__ h,
                           const float* __restrict__ a,
                           float* __restrict__ s_src,
                           float* __restrict__ s_dst, int n) {
  int node = blockIdx.x * blockDim.x + threadIdx.x;
  if (node >= n) return;
  const float* hr = h + (size_t)node * OUT_DIM;
  float ss = 0.0f, sd = 0.0f;
#pragma unroll 8
  for (int d = 0; d < OUT_DIM; ++d) {
    float v = hr[d];
    ss = fmaf(v, a[d], ss);
    sd = fmaf(v, a[OUT_DIM + d], sd);
  }
  s_src[node] = ss;
  s_dst[node] = sd;
}

// Monotonic float atomic max via sign-split int/uint atomics
// (lowers to global_atomic_max_i32 / global_atomic_min_u32; works with -inf init).
__device__ __forceinline__ void atomicMaxFloat(float* addr, float val) {
  int iv = __float_as_int(val);
  if (iv >= 0) {
    atomicMax((int*)addr, iv);
  } else {
    atomicMin((unsigned int*)addr, (unsigned int)iv);
  }
}

// ---------------------------------------------------------------------------
// Kernel 4: per-edge pre-softmax score + segment max over dst
// ---------------------------------------------------------------------------
__global__ void gat_edge_max(const long long* __restrict__ ei, int E,
                             const float* __restrict__ s_src,
                             const float* __restrict__ s_dst,
                             float* __restrict__ evals,
                             float* __restrict__ m) {
  int e = blockIdx.x * blockDim.x + threadIdx.x;
  if (e >= E) return;
  int src = (int)ei[e];
  int dst = (int)ei[(size_t)E + e];
  float v = s_src[src] + s_dst[dst];
  v = (v > 0.0f) ? v : NEG_SLOPE * v;   // LeakyReLU(0.2)
  evals[e] = v;
  atomicMaxFloat(&m[dst], v);
}

// ---------------------------------------------------------------------------
// Kernel 5: ex = exp(e - m[dst]);  denom[dst] += ex
// ---------------------------------------------------------------------------
__global__ void gat_edge_exp(const long long* __restrict__ ei, int E,
                             const float* __restrict__ m,
                             float* __restrict__ evals,
                             float* __restrict__ denom) {
  int e = blockIdx.x * blockDim.x + threadIdx.x;
  if (e >= E) return;
  int dst = (int)ei[(size_t)E + e];
  float ex = __expf(evals[e] - m[dst]);
  evals[e] = ex;
  atomicAdd(&denom[dst], ex);
}

// ---------------------------------------------------------------------------
// Kernel 6: out[dst] += h[src] * (ex / denom[dst])
// 64 threads per edge (one per output dim); 4 edges per 256-thread block.
// h/out both fit in L2 -> gathers + float atomics run at L2 rate.
// ---------------------------------------------------------------------------
__global__ void gat_scatter(const long long* __restrict__ ei, int E,
                            const float* __restrict__ h,
                            const float* __restrict__ evals,
                            const float* __restrict__ denom,
                            float* __restrict__ out) {
  int d  = threadIdx.x & (OUT_DIM - 1);
  int eo = threadIdx.x >> 6;
  int e  = blockIdx.x * 4 + eo;
  if (e >= E) return;
  int src = (int)ei[e];
  int dst = (int)ei[(size_t)E + e];
  float alpha = evals[e] / denom[dst];
  float v = h[(size_t)src * OUT_DIM + d] * alpha;
  atomicAdd(&out[(size_t)dst * OUT_DIM + d], v);
}

// ---------------------------------------------------------------------------
extern "C" void kernel_launch(void* const* d_in, const int* in_sizes, int n_in,
                              void* d_out, int out_size, void* d_ws, size_t ws_size,
                              hipStream_t stream) {
  const float* x      = (const float*)d_in[0];
  const float* W      = (const float*)d_in[1];
  const float* a      = (const float*)d_in[2];
  const long long* ei = (const long long*)d_in[3];
  // d_in[4] = num_nodes scalar (unused; derive sizes from in_sizes)

  const int N = in_sizes[0] / IN_DIM;   // 100000
  const int E = in_sizes[3] / 2;        // 1600000

  float* out = (float*)d_out;

  // Workspace layout
  float* h     = (float*)d_ws;          // N * 64
  float* s_src = h + (size_t)N * OUT_DIM;
  float* s_dst = s_src + N;
  float* m     = s_dst + N;
  float* denom = m + N;
  float* evals = denom + N;             // E floats

  // 1) GEMM: one wave per 16-row block, 8 waves (256 thr) per block
  {
    int nWaves = (N + 15) / 16;
    int blocks = (nWaves + 7) / 8;
    gat_gemm_wmma<<<blocks, 256, 0, stream>>>(x, W, h, N);
  }
  // 2) init out / m / denom
  {
    int total = N * OUT_DIM;
    gat_init<<<(total + 255) / 256, 256, 0, stream>>>(out, m, denom, N);
  }
  // 3) node scores
  gat_scores<<<(N + 255) / 256, 256, 0, stream>>>(h, a, s_src, s_dst, N);
  // 4) edge score + segment max
  gat_edge_max<<<(E + 255) / 256, 256, 0, stream>>>(ei, E, s_src, s_dst, evals, m);
  // 5) exp + segment sum
  gat_edge_exp<<<(E + 255) / 256, 256, 0, stream>>>(ei, E, m, evals, denom);
  // 6) weighted scatter
  gat_scatter<<<(E + 3) / 4, 256, 0, stream>>>(ei, E, h, evals, denom, out);
}